// SBCSGraphSAGE_26499948216591
// MI455X (gfx1250) — compile-verified
//
#include <hip/hip_runtime.h>
#include <hip/hip_bf16.h>
#include <math.h>

// ---------------------------------------------------------------------------
// GraphSAGE (3x SAGEConv + GraphNorm + GELU) + MLP head, for MI455X (gfx1250).
// All GEMMs run through V_WMMA_F32_16X16X4_F32 (full f32 precision, wave32).
// Aggregation is an atomic scatter (L2-resident working set on 192MB L2).
// ---------------------------------------------------------------------------

#define NNODES 100000
#define NEDGES 1600000
#define F_IN   20
#define HDIM   128
#define GN_EPS 1e-5f

typedef float v2f __attribute__((ext_vector_type(2)));
typedef float v8f __attribute__((ext_vector_type(8)));

// ---------------------------------------------------------------- utilities
__global__ void zero_kernel(float* __restrict__ p, int n) {
    int i = blockIdx.x * blockDim.x + threadIdx.x;
    if (i < n) p[i] = 0.0f;
}

__global__ void deg_kernel(const int* __restrict__ ei, float* __restrict__ deg) {
    int e = blockIdx.x * blockDim.x + threadIdx.x;
    if (e < NEDGES) {
        int dst = ei[NEDGES + e];
        atomicAdd(&deg[dst], 1.0f);
    }
}

// one thread handles 4 consecutive features of one edge; for DIN=128 one
// wave32 covers a whole edge (coalesced 512B gather + coalesced atomics).
template <int DIN>
__global__ void scatter_add_kernel(const int* __restrict__ ei,
                                   const float* __restrict__ hin,
                                   float* __restrict__ agg) {
    constexpr int K4 = DIN / 4;
    int idx = blockIdx.x * blockDim.x + threadIdx.x;
    if (idx >= NEDGES * K4) return;
    int e = idx / K4;
    int f = (idx % K4) * 4;
    int src = ei[e];
    int dst = ei[NEDGES + e];
    float4 v = *(const float4*)&hin[(size_t)src * DIN + f];
    float* p = &agg[(size_t)dst * DIN + f];
    atomicAdd(p + 0, v.x);
    atomicAdd(p + 1, v.y);
    atomicAdd(p + 2, v.z);
    atomicAdd(p + 3, v.w);
}

// ------------------------------------------------- fused SAGE linear (WMMA)
// out[n,j] = sum_k aggscaled[n,k]*Wl[j,k] + sum_k hin[n,k]*Wr[j,k] + bl[j]
// Treated as one GEMM with A = [agg*inv(deg) | hin]  (N x K, K = 2*DIN).
// Block = 256 threads = 8 waves; block computes a 16-row x 128-col tile,
// wave w owns columns [16w, 16w+16). f32 WMMA 16x16x4, K-loop step 4.
template <int K, int DIN>
__global__ __launch_bounds__(256) void sage_gemm_kernel(
    const float* __restrict__ agg, const float* __restrict__ hin,
    const float* __restrict__ deg, const float* __restrict__ Wl,
    const float* __restrict__ Wr, const float* __restrict__ bias,
    float* __restrict__ out) {
    constexpr int LDK = K + 4;
    __shared__ float Alds[16 * LDK];
    __shared__ float invs[16];

    const int rowBase = blockIdx.x * 16;
    const int tid = threadIdx.x;

    if (tid < 16) {
        float d = deg[rowBase + tid];
        invs[tid] = 1.0f / fmaxf(d, 1.0f);
    }
    __syncthreads();

    // stage A tile: [agg*inv | hin], 16 rows x K, float4 granularity
    constexpr int K4 = K / 4;
    for (int idx = tid; idx < 16 * K4; idx += 256) {
        int m = idx / K4;
        int k = (idx % K4) * 4;
        float4 v;
        if (k < DIN) {
            v = *(const float4*)&agg[(size_t)(rowBase + m) * DIN + k];
            float s = invs[m];
            v.x *= s; v.y *= s; v.z *= s; v.w *= s;
        } else {
            v = *(const float4*)&hin[(size_t)(rowBase + m) * DIN + (k - DIN)];
        }
        *(float4*)&Alds[m * LDK + k] = v;
    }
    __syncthreads();

    const int wave  = tid >> 5;   // 0..7
    const int lane  = tid & 31;
    const int half  = lane >> 4;  // 0: lanes 0-15, 1: lanes 16-31
    const int idx16 = lane & 15;
    const int jcol  = wave * 16 + idx16;

    float bsc = bias[jcol];
    v8f acc = {bsc, bsc, bsc, bsc, bsc, bsc, bsc, bsc};

#pragma unroll
    for (int kb = 0; kb < K; kb += 4) {
        const int ka = kb + 2 * half;  // A layout: vgpr0->k={0,2}, vgpr1->k={1,3}
        v2f a, b;
        a.x = Alds[idx16 * LDK + ka];
        a.y = Alds[idx16 * LDK + ka + 1];
        // kb is a compile-time constant after unroll; (kb<DIN) folds, and
        // kb<DIN implies kb+3<DIN because 4|DIN.
        const float* Wsel = (kb < DIN) ? Wl : Wr;
        const int kk = (kb < DIN) ? ka : (ka - DIN);
        b.x = Wsel[jcol * DIN + kk];
        b.y = Wsel[jcol * DIN + kk + 1];
        acc = __builtin_amdgcn_wmma_f32_16x16x4_f32(
            false, a, false, b, (short)0, acc, false, false);
    }

#pragma unroll
    for (int r = 0; r < 8; ++r) {
        int m = r + 8 * half;  // C layout: lanes16-31 hold M=r+8
        out[(size_t)(rowBase + m) * HDIM + jcol] = acc[r];
    }
}

// ------------------------------------------------------ head GEMM (WMMA+GELU)
template <int K, int NCOL>
__global__ __launch_bounds__(NCOL * 2) void head_gemm_kernel(
    const float* __restrict__ A, const float* __restrict__ W,
    const float* __restrict__ bias, float* __restrict__ out) {
    constexpr int LDK = K + 4;
    constexpr int NTHREADS = NCOL * 2;  // (NCOL/16) waves * 32
    __shared__ float Alds[16 * LDK];

    const int rowBase = blockIdx.x * 16;
    const int tid = threadIdx.x;

    constexpr int K4 = K / 4;
    for (int idx = tid; idx < 16 * K4; idx += NTHREADS) {
        int m = idx / K4;
        int k = (idx % K4) * 4;
        *(float4*)&Alds[m * LDK + k] =
            *(const float4*)&A[(size_t)(rowBase + m) * K + k];
    }
    __syncthreads();

    const int wave  = tid >> 5;
    const int lane  = tid & 31;
    const int half  = lane >> 4;
    const int idx16 = lane & 15;
    const int jcol  = wave * 16 + idx16;

    float bsc = bias[jcol];
    v8f acc = {bsc, bsc, bsc, bsc, bsc, bsc, bsc, bsc};

#pragma unroll
    for (int kb = 0; kb < K; kb += 4) {
        const int ka = kb + 2 * half;
        v2f a, b;
        a.x = Alds[idx16 * LDK + ka];
        a.y = Alds[idx16 * LDK + ka + 1];
        b.x = W[jcol * K + ka];
        b.y = W[jcol * K + ka + 1];
        acc = __builtin_amdgcn_wmma_f32_16x16x4_f32(
            false, a, false, b, (short)0, acc, false, false);
    }

#pragma unroll
    for (int r = 0; r < 8; ++r) {
        int m = r + 8 * half;
        float y = acc[r];
        float g = 0.5f * y * (1.0f + erff(y * 0.70710678118654752f));  // exact GELU
        out[(size_t)(rowBase + m) * NCOL + jcol] = g;
    }
}

// ------------------------------------------------------------- GraphNorm stats
// stats[c] = sum over nodes, stats[128+c] = sum of squares
__global__ __launch_bounds__(128) void stats_kernel(const float* __restrict__ x,
                                                    float* __restrict__ stats) {
    const int c = threadIdx.x;               // channel 0..127
    const int rpb = (NNODES + gridDim.x - 1) / gridDim.x;
    const int r0 = blockIdx.x * rpb;
    const int r1 = min(r0 + rpb, NNODES);
    float s = 0.0f, s2 = 0.0f;
    for (int r = r0; r < r1; ++r) {
        float v = x[(size_t)r * HDIM + c];
        s += v;
        s2 += v * v;
    }
    atomicAdd(&stats[c], s);
    atomicAdd(&stats[HDIM + c], s2);
}

// in-place GraphNorm + exact GELU.
// var = E[(x-am)^2] = E[x^2] + m^2 * a * (a-2)   (since E[x]=m)
__global__ void norm_gelu_kernel(float* __restrict__ x,
                                 const float* __restrict__ stats,
                                 const float* __restrict__ w,
                                 const float* __restrict__ b,
                                 const float* __restrict__ a) {
    int idx = blockIdx.x * blockDim.x + threadIdx.x;
    if (idx >= NNODES * HDIM) return;
    int c = idx & (HDIM - 1);
    const float invN = 1.0f / (float)NNODES;
    float m  = stats[c] * invN;
    float aa = a[c];
    float var = stats[HDIM + c] * invN + m * m * aa * (aa - 2.0f);
    float cv = x[idx] - aa * m;
    float y  = w[c] * cv * rsqrtf(var + GN_EPS) + b[c];
    x[idx]   = 0.5f * y * (1.0f + erff(y * 0.70710678118654752f));
}

// ----------------------------------------------------------- final 32->1 layer
__global__ void head_final_kernel(const float* __restrict__ t32,
                                  const float* __restrict__ W,
                                  const float* __restrict__ b,
                                  float* __restrict__ out) {
    int n = blockIdx.x * blockDim.x + threadIdx.x;
    if (n >= NNODES) return;
    float s = b[0];
#pragma unroll
    for (int k = 0; k < 32; ++k) s += t32[(size_t)n * 32 + k] * W[k];
    out[n] = 1.0f / (1.0f + expf(-s));
}

// ---------------------------------------------------------------------------
extern "C" void kernel_launch(void* const* d_in, const int* in_sizes, int n_in,
                              void* d_out, int out_size, void* d_ws,
                              size_t ws_size, hipStream_t stream) {
    (void)in_sizes; (void)n_in; (void)out_size; (void)ws_size;

    const float* x  = (const float*)d_in[0];
    const int*   ei = (const int*)d_in[1];
    const float* Wl[3] = {(const float*)d_in[2], (const float*)d_in[8],  (const float*)d_in[14]};
    const float* bl[3] = {(const float*)d_in[3], (const float*)d_in[9],  (const float*)d_in[15]};
    const float* Wr[3] = {(const float*)d_in[4], (const float*)d_in[10], (const float*)d_in[16]};
    const float* gw[3] = {(const float*)d_in[5], (const float*)d_in[11], (const float*)d_in[17]};
    const float* gb[3] = {(const float*)d_in[6], (const float*)d_in[12], (const float*)d_in[18]};
    const float* ga[3] = {(const float*)d_in[7], (const float*)d_in[13], (const float*)d_in[19]};
    const float* hW1 = (const float*)d_in[20];
    const float* hb1 = (const float*)d_in[21];
    const float* hW2 = (const float*)d_in[22];
    const float* hb2 = (const float*)d_in[23];
    const float* hW3 = (const float*)d_in[24];
    const float* hb3 = (const float*)d_in[25];

    // workspace layout (floats); ~154 MB total
    float* ws = (float*)d_ws;
    const size_t NH = (size_t)NNODES * HDIM;  // 12.8M
    float* deg   = ws;                        // N
    float* agg   = ws + 102400;               // N*128
    float* bufA  = agg + NH;                  // N*128
    float* bufB  = bufA + NH;                 // N*128
    float* stats = bufB + NH;                 // 256
    float* t64   = agg;                       // reuse agg after layer 2
    float* t32   = agg + (size_t)NNODES * 64;

    const int rowBlocks = NNODES / 16;  // 6250 exactly

    // ---- degrees (once) ----
    zero_kernel<<<(NNODES + 255) / 256, 256, 0, stream>>>(deg, NNODES);
    deg_kernel<<<(NEDGES + 255) / 256, 256, 0, stream>>>(ei, deg);

    // ---- layer 0: F_IN=20 -> 128 ----
    zero_kernel<<<(NNODES * F_IN + 255) / 256, 256, 0, stream>>>(agg, NNODES * F_IN);
    scatter_add_kernel<F_IN>
        <<<(NEDGES * (F_IN / 4) + 255) / 256, 256, 0, stream>>>(ei, x, agg);
    sage_gemm_kernel<2 * F_IN, F_IN>
        <<<rowBlocks, 256, 0, stream>>>(agg, x, deg, Wl[0], Wr[0], bl[0], bufA);
    zero_kernel<<<1, 256, 0, stream>>>(stats, 256);
    stats_kernel<<<256, 128, 0, stream>>>(bufA, stats);
    norm_gelu_kernel<<<(NNODES * HDIM + 255) / 256, 256, 0, stream>>>(
        bufA, stats, gw[0], gb[0], ga[0]);

    // ---- layers 1,2: 128 -> 128 ----
    for (int L = 1; L <= 2; ++L) {
        const float* hin = (L == 1) ? bufA : bufB;
        float* hout      = (L == 1) ? bufB : bufA;
        zero_kernel<<<(int)((NH + 255) / 256), 256, 0, stream>>>(agg, (int)NH);
        scatter_add_kernel<HDIM>
            <<<(NEDGES * (HDIM / 4) + 255) / 256, 256, 0, stream>>>(ei, hin, agg);
        sage_gemm_kernel<2 * HDIM, HDIM>
            <<<rowBlocks, 256, 0, stream>>>(agg, hin, deg, Wl[L], Wr[L], bl[L], hout);
        zero_kernel<<<1, 256, 0, stream>>>(stats, 256);
        stats_kernel<<<256, 128, 0, stream>>>(hout, stats);
        norm_gelu_kernel<<<(NNODES * HDIM + 255) / 256, 256, 0, stream>>>(
            hout, stats, gw[L], gb[L], ga[L]);
    }

    // ---- MLP head: 128 -> 64 -> 32 -> 1 (+sigmoid) ----
    head_gemm_kernel<128, 64><<<rowBlocks, 128, 0, stream>>>(bufA, hW1, hb1, t64);
    head_gemm_kernel<64, 32><<<rowBlocks, 64, 0, stream>>>(t64, hW2, hb2, t32);
    head_final_kernel<<<(NNODES + 255) / 256, 256, 0, stream>>>(
        t32, hW3, hb3, (float*)d_out);
}